// S4_59742995087662
// MI455X (gfx1250) — compile-verified
//
#include <hip/hip_runtime.h>
#include <math.h>

// ---------------------------------------------------------------------------
// S4D forward on MI455X (gfx1250), chunked state-space (SSD) formulation.
//   x:(L,B,H) fp32 -> y:(L,B,H) fp32
//   L=4096, B=8, H=1024, N=16 modes, chunk Q=64.
// intra-chunk: 64x64 lower-tri Toeplitz GEMM via V_WMMA_F32_16X16X4_F32
//   (all 4 waves run the full K=16-tile loop; above-diagonal A fragments are
//    structurally zero, keeping trip counts compile-time so fragments stay in
//    statically-indexed VGPRs -- no v_movrels/M0 indexing)
// inter-chunk: complex diagonal state recurrence (VALU), exact math.
// Memory-bound: 256 MB total traffic ~ 11 us at 23.3 TB/s HBM; x (128 MB)
// fits in 192 MB L2 so the per-channel 4B gathers cost one HBM pass.
// ---------------------------------------------------------------------------

#define Hdim  1024
#define Ndim  16
#define Lseq  4096
#define NBat  8
#define UPAD  16      // ulds/ylds padded to 16 columns (zero pad for WMMA N)
#define Q     64
#define NCHUNK (Lseq / Q)
#define SCALE_F 0.25f   // sqrt(1/16)

typedef float v2f __attribute__((ext_vector_type(2)));
typedef float v8f __attribute__((ext_vector_type(8)));

__device__ __forceinline__ void cmul(float ar, float ai, float br, float bi,
                                     float& cr, float& ci) {
    cr = ar * br - ai * bi;
    ci = ar * bi + ai * br;
}

// integer power of a complex number via binary exponentiation (few-ulp exact)
__device__ __forceinline__ void cpow_int(float ar, float ai, int p,
                                         float& rr, float& ri) {
    float xr = ar, xi = ai;
    rr = 1.0f; ri = 0.0f;
    while (p) {
        if (p & 1) { float tr, ti; cmul(rr, ri, xr, xi, tr, ti); rr = tr; ri = ti; }
        p >>= 1;
        if (p)     { float tr, ti; cmul(xr, xi, xr, xi, tr, ti); xr = tr; xi = ti; }
    }
}

// ---------------------------------------------------------------------------
// Setup: per (h,n) compute bilinear-discretized mode:
//   dA  = (1+dtA/2)/(1-dtA/2),  cb2 = 2*SCALE * C*B*dt/(1-dtA/2),  dAQ = dA^64
// ws layout (floats): [dA_re | dA_im | cb_re | cb_im | dQ_re | dQ_im], each H*N
// ---------------------------------------------------------------------------
__global__ void s4d_setup(const float* __restrict__ log_dt,
                          const float* __restrict__ log_A_real,
                          const float* __restrict__ A_imag,
                          const float* __restrict__ Bp,
                          const float* __restrict__ Cp,
                          float* __restrict__ ws) {
    int idx = blockIdx.x * blockDim.x + threadIdx.x;
    if (idx >= Hdim * Ndim) return;
    int h = idx >> 4;

    float dt  = expf(log_dt[h]);
    float Are = -expf(log_A_real[idx]);
    float Aim = A_imag[idx];

    float zr = 0.5f * dt * Are, zi = 0.5f * dt * Aim;   // dtA/2
    float nr = 1.0f + zr, ni = zi;                      // numerator
    float dr = 1.0f - zr, di = -zi;                     // denominator
    float inv = 1.0f / (dr * dr + di * di);
    float dAr = (nr * dr + ni * di) * inv;
    float dAi = (ni * dr - nr * di) * inv;

    float Br = Bp[2 * idx], Bi = Bp[2 * idx + 1];
    float Cr = Cp[2 * idx], Ci = Cp[2 * idx + 1];
    float tr, ti;
    cmul(Cr, Ci, Br, Bi, tr, ti);
    tr *= dt; ti *= dt;
    float cbr = (tr * dr + ti * di) * inv;              // /(1-dtA/2)
    float cbi = (ti * dr - tr * di) * inv;
    cbr *= 2.0f * SCALE_F;
    cbi *= 2.0f * SCALE_F;

    float qr = dAr, qi = dAi;                           // dA^64 by 6 squarings
    #pragma unroll
    for (int s = 0; s < 6; ++s) { float ur, ui; cmul(qr, qi, qr, qi, ur, ui); qr = ur; qi = ui; }

    const int HN = Hdim * Ndim;
    ws[0 * HN + idx] = dAr;
    ws[1 * HN + idx] = dAi;
    ws[2 * HN + idx] = cbr;
    ws[3 * HN + idx] = cbi;
    ws[4 * HN + idx] = qr;
    ws[5 * HN + idx] = qi;
}

// ---------------------------------------------------------------------------
// Main: one 128-thread block (4 wave32) per channel h, sequential over chunks.
// ---------------------------------------------------------------------------
__global__ __launch_bounds__(128) void s4d_main(const float* __restrict__ x,
                                                const float* __restrict__ Dvec,
                                                const float* __restrict__ ws,
                                                float* __restrict__ out) {
    const int h   = blockIdx.x;
    const int tid = threadIdx.x;

    __shared__ float sdA_re[Ndim], sdA_im[Ndim];
    __shared__ float scb_re[Ndim], scb_im[Ndim];
    __shared__ float sdQ_re[Ndim], sdQ_im[Ndim];
    __shared__ float E_re[Q][Ndim], E_im[Q][Ndim];   // cb2 * dA^{i+1}
    __shared__ float W_re[Q][Ndim], W_im[Q][Ndim];   // dA^{63-j}
    __shared__ float klds[Q];                        // local taps k[0..63]
    __shared__ float ulds[Q][UPAD];                  // u chunk (cols 8..15 = 0)
    __shared__ float ylds[Q][UPAD];                  // intra result
    __shared__ float S_re[NBat][Ndim], S_im[NBat][Ndim];  // carried states

    const int HN = Hdim * Ndim;
    if (tid < Ndim) {
        sdA_re[tid] = ws[0 * HN + h * Ndim + tid];
        sdA_im[tid] = ws[1 * HN + h * Ndim + tid];
        scb_re[tid] = ws[2 * HN + h * Ndim + tid];
        scb_im[tid] = ws[3 * HN + h * Ndim + tid];
        sdQ_re[tid] = ws[4 * HN + h * Ndim + tid];
        sdQ_im[tid] = ws[5 * HN + h * Ndim + tid];
    }
    { S_re[tid >> 4][tid & 15] = 0.0f; S_im[tid >> 4][tid & 15] = 0.0f; }
    // zero the N-padding columns of ulds once (never written again)
    #pragma unroll
    for (int r = 0; r < 4; ++r) {
        int idx = tid + r * 128;                 // 512 pad slots
        ulds[idx >> 3][(idx & 7) + NBat] = 0.0f;
    }
    __syncthreads();

    // Per-channel coefficient tables (once per block, reused for 64 chunks)
    if (tid < Q) {
        float kr = 0.0f;
        for (int n = 0; n < Ndim; ++n) {
            float dr = sdA_re[n], di = sdA_im[n];
            float cr = scb_re[n], ci = scb_im[n];
            float pr, pi; cpow_int(dr, di, tid, pr, pi);        // dA^i
            kr += cr * pr - ci * pi;                            // Re(cb2*dA^i)
            float qr, qi; cmul(pr, pi, dr, di, qr, qi);         // dA^{i+1}
            E_re[tid][n] = cr * qr - ci * qi;
            E_im[tid][n] = cr * qi + ci * qr;
            float wr, wi; cpow_int(dr, di, 63 - tid, wr, wi);   // dA^{63-j}
            W_re[tid][n] = wr;
            W_im[tid][n] = wi;
        }
        klds[tid] = kr;
    }
    __syncthreads();

    // WMMA A-fragments for this wave's M tile: T[i][j] = k[i-j] (lower tri).
    // f32 16x16x4 A layout: lanes 0-15 hold K={0,1}, lanes 16-31 hold K={2,3}.
    // All 16 K-tiles built for every wave (above-diagonal ones are all zero)
    // so the WMMA loop below has a constant trip count -> static reg indexing.
    const int lane = tid & 31;
    const int mt   = tid >> 5;            // wave id == M tile (0..3)
    const int mrow = mt * 16 + (lane & 15);
    const int kb   = (lane >> 4) * 2;
    v2f afrag[16];
    #pragma unroll
    for (int kt = 0; kt < 16; ++kt) {
        int c0 = kt * 4 + kb;
        afrag[kt].x = (c0     <= mrow) ? klds[mrow - c0]     : 0.0f;
        afrag[kt].y = (c0 + 1 <= mrow) ? klds[mrow - c0 - 1] : 0.0f;
    }

    const float Dh = Dvec[h];
    const int bb = tid >> 4;              // state-update mapping (b,n)
    const int nn = tid & 15;
    const int nidx = lane & 15;           // WMMA B/N column

    for (int c = 0; c < NCHUNK; ++c) {
        __syncthreads();
        // ---- load u chunk (and prefetch next chunk into L2) ----
        const int base = c * Q * (NBat * Hdim);
        #pragma unroll
        for (int r = 0; r < 4; ++r) {
            int idx = tid + r * 128;
            int j = idx >> 3, b = idx & 7;
            int g = base + j * (NBat * Hdim) + b * Hdim + h;
            ulds[j][b] = x[g];
            if (c + 1 < NCHUNK) __builtin_prefetch(&x[g + Q * NBat * Hdim], 0, 0);
        }
        __syncthreads();

        // ---- intra-chunk: Y = T(64x64, lower-tri Toeplitz) * U(64x16pad) ----
        v8f acc = {};
        #pragma unroll
        for (int kt = 0; kt < 16; ++kt) {
            v2f bfrag;
            int r0 = kt * 4 + kb;
            bfrag.x = ulds[r0][nidx];
            bfrag.y = ulds[r0 + 1][nidx];
            acc = __builtin_amdgcn_wmma_f32_16x16x4_f32(
                false, afrag[kt], false, bfrag, (short)0, acc, false, false);
        }
        {   // D layout: vgpr v -> row v (lanes 0-15) / v+8 (lanes 16-31), col=lane%16
            int rbase = mt * 16 + (lane >> 4) * 8;
            #pragma unroll
            for (int v = 0; v < 8; ++v) ylds[rbase + v][nidx] = acc[v];
        }
        __syncthreads();

        // ---- output: intra + inter(states) + skip ----
        #pragma unroll
        for (int r = 0; r < 4; ++r) {
            int idx = tid + r * 128;
            int i = idx >> 3, b = idx & 7;
            float y = ylds[i][b] + Dh * ulds[i][b];
            float s = 0.0f;
            #pragma unroll
            for (int n = 0; n < Ndim; ++n)
                s += E_re[i][n] * S_re[b][n] - E_im[i][n] * S_im[b][n];
            out[base + i * (NBat * Hdim) + b * Hdim + h] = y + s;
        }
        __syncthreads();

        // ---- state update: S' = dA^Q * S + sum_j dA^{63-j} u[j] ----
        {
            float ar = 0.0f, ai = 0.0f;
            for (int j = 0; j < Q; ++j) {
                float u = ulds[j][bb];
                ar += W_re[j][nn] * u;
                ai += W_im[j][nn] * u;
            }
            float sr = S_re[bb][nn], si = S_im[bb][nn];
            float qr = sdQ_re[nn],  qi = sdQ_im[nn];
            S_re[bb][nn] = qr * sr - qi * si + ar;
            S_im[bb][nn] = qr * si + qi * sr + ai;
        }
    }
}

extern "C" void kernel_launch(void* const* d_in, const int* in_sizes, int n_in,
                              void* d_out, int out_size, void* d_ws, size_t ws_size,
                              hipStream_t stream) {
    (void)in_sizes; (void)n_in; (void)out_size; (void)ws_size;
    const float* x          = (const float*)d_in[0];
    const float* log_dt     = (const float*)d_in[1];
    const float* log_A_real = (const float*)d_in[2];
    const float* A_imag     = (const float*)d_in[3];
    const float* Bp         = (const float*)d_in[4];
    const float* Cp         = (const float*)d_in[5];
    const float* Dv         = (const float*)d_in[6];
    float* out = (float*)d_out;
    float* ws  = (float*)d_ws;   // 6 * H * N floats = 384 KB

    s4d_setup<<<(Hdim * Ndim + 255) / 256, 256, 0, stream>>>(
        log_dt, log_A_real, A_imag, Bp, Cp, ws);
    s4d_main<<<Hdim, 128, 0, stream>>>(x, Dv, ws, out);
}